// DynamicQAgent_11639361372465
// MI455X (gfx1250) — compile-verified
//
#include <hip/hip_runtime.h>
#include <math.h>
#include <stdint.h>

// Problem constants (fixed by the harness/reference).
#define B_SESS 4096
#define T_STEPS 1000
#define TC 40                    // timesteps per staged chunk
#define NC (T_STEPS / TC)        // 25 chunks (exact)
#define ROWS 32                  // sessions per block == blockDim.x (one wave32)
#define ROW_F (TC * 3)           // 120 floats per session per chunk
#define ROW_PAD_F 124            // padded row stride: 31 x 16B units (16B aligned)
#define UNITS_PER_ROW (ROW_F / 4)        // 30 x 16B units per row
#define ROUNDS UNITS_PER_ROW             // 30 async b128 issues per wave per chunk
#define ROW_BYTES_GLOBAL (T_STEPS * 3 * 4)  // 12000 bytes per session in global

__global__ __launch_bounds__(ROWS)
void DynamicQAgent_scan_kernel(const float* __restrict__ inp,
                               const float* __restrict__ alpha_logits,
                               const float* __restrict__ gamma_logits,
                               const float* __restrict__ k_vals,
                               float2* __restrict__ out)
{
    // Double-buffered input staging: 2 x 32 sessions x 124 floats = 31.7 KB.
    __shared__ alignas(16) float sh[2][ROWS][ROW_PAD_F];

    const int tid   = threadIdx.x;              // session within block, 0..31
    const int blk   = blockIdx.x;               // 0..127
    const int gsess = blk * ROWS + tid;         // global session id

    // Per-block global base; per-lane 32-bit byte offsets ride in the VADDR
    // VGPR of the GVS-mode async copy (SADDR = this pointer in an SGPR pair).
    const float* gbase = inp + (size_t)blk * ROWS * (T_STEPS * 3);

    const uint32_t lds_base[2] = {
        (uint32_t)(uintptr_t)(&sh[0][0][0]),
        (uint32_t)(uintptr_t)(&sh[1][0][0])
    };

    // Issue the async DMA of one 40-step chunk for all 32 sessions of this wave.
    // 30 x (32 lanes x 16B) = 15 KB per chunk; lanes walk consecutive 16B units
    // so each instruction covers mostly-contiguous 480B session-row segments.
    auto issue_chunk = [&](int c, uint32_t ldsb) {
        const uint32_t chunk_byte = (uint32_t)(c * TC * 3 * 4);   // 480*c, 16B-aligned
        #pragma unroll
        for (int r = 0; r < ROUNDS; ++r) {
            uint32_t u     = (uint32_t)(r * ROWS + tid);          // 0..959
            uint32_t srow  = u / UNITS_PER_ROW;                   // session row 0..31
            uint32_t kk    = u - srow * UNITS_PER_ROW;            // 16B unit in row
            uint32_t goff  = srow * (uint32_t)ROW_BYTES_GLOBAL + chunk_byte + kk * 16u;
            uint32_t laddr = ldsb + srow * (uint32_t)(ROW_PAD_F * 4) + kk * 16u;
            asm volatile("global_load_async_to_lds_b128 %0, %1, %2"
                         :
                         : "v"(laddr), "v"(goff), "s"(gbase)
                         : "memory");
        }
    };

    // Model parameters: uniform scalar loads + sigmoid, once per thread.
    const float a0 = 1.0f / (1.0f + expf(-alpha_logits[0]));
    const float a1 = 1.0f / (1.0f + expf(-alpha_logits[1]));
    const float a2 = 1.0f / (1.0f + expf(-alpha_logits[2]));
    const float a3 = 1.0f / (1.0f + expf(-alpha_logits[3]));
    const float g0 = 1.0f / (1.0f + expf(-gamma_logits[0]));
    const float g1 = 1.0f / (1.0f + expf(-gamma_logits[1]));
    const float k0 = k_vals[0], k1 = k_vals[1], k2 = k_vals[2], k3 = k_vals[3];

    float s0 = 0.0f, s1 = 0.0f;          // state
    float lam0 = 0.5f, lam1 = 0.5f;      // learning rates

    // Prime the pipeline with chunk 0.
    issue_chunk(0, lds_base[0]);

    for (int c = 0; c < NC; ++c) {
        if (c + 1 < NC) {
            issue_chunk(c + 1, lds_base[(c + 1) & 1]);
            // <=60 outstanding; in-order completion => counter<=30 means all of
            // chunk c has landed in LDS (the newest 30 belong to chunk c+1).
            asm volatile("s_wait_asynccnt 0x1e" ::: "memory");
        } else {
            asm volatile("s_wait_asynccnt 0x0" ::: "memory");
        }
        // Single-wave workgroup: the consuming wave is the issuing wave, so no
        // barrier is required for LDS visibility.

        const float* row = &sh[c & 1][tid][0];
        const int t0 = c * TC;

        #pragma unroll 4
        for (int tt = 0; tt < TC; ++tt) {
            const float cL = row[tt * 3 + 0];
            const float cR = row[tt * 3 + 1];
            const float o  = row[tt * 3 + 2];
            const float om = 1.0f - o;

            // sel_L = [cL*o, cL*(1-o), cR*o, cR*(1-o)]; sel_R = sel_L[perm], perm=[2,3,0,1]
            const float sl0 = cL * o, sl1 = cL * om, sl2 = cR * o, sl3 = cR * om;

            const float dL0 = k0 - s0, dL1 = k1 - s0, dL2 = k2 - s0, dL3 = k3 - s0;
            const float dR0 = k0 - s1, dR1 = k1 - s1, dR2 = k2 - s1, dR3 = k3 - s1;

            const float sumL = sl0 * dL0 * a0 + sl1 * dL1 * a1
                             + sl2 * dL2 * a2 + sl3 * dL3 * a3;
            const float sumR = sl2 * dR0 * a0 + sl3 * dR1 * a1
                             + sl0 * dR2 * a2 + sl1 * dR3 * a3;

            const float ns0 = fmaf(sumL, lam0, s0);
            const float ns1 = fmaf(sumR, lam1, s1);

            // gdiffs = [dL0, dL1, dR0, dR1], gammas = [g0, g1, g0, g1]; uses OLD lam0.
            const float ld = (fabsf(dL0) - lam0) * sl0 * g0
                           + (fabsf(dL1) - lam0) * sl1 * g1
                           + (fabsf(dR0) - lam0) * sl2 * g0
                           + (fabsf(dR1) - lam0) * sl3 * g1;

            lam0 += ld;
            lam1 += ld;
            s0 = ns0;
            s1 = ns1;

            out[(size_t)gsess * T_STEPS + (t0 + tt)] = make_float2(ns0, ns1);
        }
    }
}

extern "C" void kernel_launch(void* const* d_in, const int* in_sizes, int n_in,
                              void* d_out, int out_size, void* d_ws, size_t ws_size,
                              hipStream_t stream) {
    (void)in_sizes; (void)n_in; (void)out_size; (void)d_ws; (void)ws_size;
    const float* inp = (const float*)d_in[0];   // (B, T, 3) f32
    const float* al  = (const float*)d_in[1];   // (4,)  alpha logits
    const float* gl  = (const float*)d_in[2];   // (2,)  gamma logits
    const float* kv  = (const float*)d_in[3];   // (4,)  k values
    float2* out = (float2*)d_out;               // (B, T, 2) f32

    dim3 grid(B_SESS / ROWS);   // 128 single-wave workgroups spread across WGPs
    dim3 block(ROWS);           // 32 = one wave32
    hipLaunchKernelGGL(DynamicQAgent_scan_kernel, grid, block, 0, stream,
                       inp, al, gl, kv, out);
}